// DTWLoss_635655160680
// MI455X (gfx1250) — compile-verified
//
#include <hip/hip_runtime.h>
#include <cstdint>

typedef __attribute__((ext_vector_type(2))) float v2f;
typedef __attribute__((ext_vector_type(8))) float v8f;

#define BATCH 64
#define SEQ 512
#define FEAT 256
#define GAMMA_F 0.1f
#define INV_GAMMA 10.0f
#define BIGV 1.0e8f

#define BT 128   // output tile per workgroup (BT x BT)
#define KT 16    // K staged in LDS per outer iteration

// ---------------- row norms: |x_i|^2 per row, one wave per row ----------------
__global__ __launch_bounds__(256) void dtw_norms_kernel(
    const float* __restrict__ x, const float* __restrict__ y,
    float* __restrict__ nx, float* __restrict__ ny) {
  int gid  = blockIdx.x * blockDim.x + threadIdx.x;
  int wave = gid >> 5;
  int lane = gid & 31;
  const int ROWS = BATCH * SEQ;
  if (wave >= 2 * ROWS) return;
  const float* src = (wave < ROWS) ? x : y;
  float* dst       = (wave < ROWS) ? nx : ny;
  int row          = (wave < ROWS) ? wave : wave - ROWS;
  const float4* p = (const float4*)(src + (size_t)row * FEAT);
  float4 a = p[lane];        // k = lane*4 .. +3
  float4 b = p[lane + 32];   // k = 128 + lane*4 .. +3
  float s = a.x * a.x + a.y * a.y + a.z * a.z + a.w * a.w
          + b.x * b.x + b.y * b.y + b.z * b.z + b.w * b.w;
#pragma unroll
  for (int off = 16; off > 0; off >>= 1) s += __shfl_xor(s, off, 32);
  if (lane == 0) dst[row] = s;
}

// ------- squared-distance via fp32 WMMA: D = nA + nB - 2 * A * B^T -------
// 256 threads = 8 waves; each wave owns a 32x64 sub-tile (2x4 of 16x16 WMMA).
__global__ __launch_bounds__(256) void sqdist_wmma_kernel(
    const float* __restrict__ A, const float* __restrict__ Bmat,
    const float* __restrict__ nA, const float* __restrict__ nB,
    float* __restrict__ D) {
  __shared__ float As[BT][KT + 1];
  __shared__ float Bs[BT][KT + 1];

  const int b  = blockIdx.z;
  const int i0 = blockIdx.y * BT;
  const int j0 = blockIdx.x * BT;
  const float* Ab = A    + (size_t)b * SEQ * FEAT;
  const float* Bb = Bmat + (size_t)b * SEQ * FEAT;

  const int tid  = threadIdx.x;
  const int wave = tid >> 5;
  const int lane = tid & 31;
  const int p = lane >> 4;          // half-wave: 0 -> K pair {0,1}, 1 -> {2,3}
  const int q = lane & 15;          // row/col within 16-tile
  const int wm = (wave & 3) * 32;   // wave row offset inside block tile
  const int wn = (wave >> 2) * 64;  // wave col offset inside block tile

  v8f acc[2][4];
#pragma unroll
  for (int ti = 0; ti < 2; ++ti)
#pragma unroll
    for (int tj = 0; tj < 4; ++tj) {
      v8f z = {0.f, 0.f, 0.f, 0.f, 0.f, 0.f, 0.f, 0.f};
      acc[ti][tj] = z;
    }

  const int quad = tid & 3;   // which K-quad this thread loads
  const int lr   = tid >> 2;  // 0..63, row pair (lr, lr+64)

  for (int k0 = 0; k0 < FEAT; k0 += KT) {
#pragma unroll
    for (int rr = 0; rr < 2; ++rr) {
      int r = lr + rr * 64;
      float4 va = *(const float4*)(Ab + (size_t)(i0 + r) * FEAT + k0 + quad * 4);
      As[r][quad * 4 + 0] = va.x; As[r][quad * 4 + 1] = va.y;
      As[r][quad * 4 + 2] = va.z; As[r][quad * 4 + 3] = va.w;
      float4 vb = *(const float4*)(Bb + (size_t)(j0 + r) * FEAT + k0 + quad * 4);
      Bs[r][quad * 4 + 0] = vb.x; Bs[r][quad * 4 + 1] = vb.y;
      Bs[r][quad * 4 + 2] = vb.z; Bs[r][quad * 4 + 3] = vb.w;
    }
    __syncthreads();
#pragma unroll
    for (int kk = 0; kk < KT; kk += 4) {
      v2f af[2], bf[4];
#pragma unroll
      for (int ti = 0; ti < 2; ++ti) {
        const float* ap = &As[wm + ti * 16 + q][kk + 2 * p];
        af[ti].x = ap[0]; af[ti].y = ap[1];
      }
#pragma unroll
      for (int tj = 0; tj < 4; ++tj) {
        const float* bp = &Bs[wn + tj * 16 + q][kk + 2 * p];
        bf[tj].x = bp[0]; bf[tj].y = bp[1];
      }
#pragma unroll
      for (int ti = 0; ti < 2; ++ti)
#pragma unroll
        for (int tj = 0; tj < 4; ++tj)
          acc[ti][tj] = __builtin_amdgcn_wmma_f32_16x16x4_f32(
              false, af[ti], false, bf[tj], (short)0, acc[ti][tj], false, false);
    }
    __syncthreads();
  }

  float* Dp = D + (size_t)b * SEQ * SEQ;
  const float* nAp = nA + b * SEQ;
  const float* nBp = nB + b * SEQ;
#pragma unroll
  for (int ti = 0; ti < 2; ++ti) {
#pragma unroll
    for (int tj = 0; tj < 4; ++tj) {
      int gj = j0 + wn + tj * 16 + q;
      float nbv = nBp[gj];
#pragma unroll
      for (int e = 0; e < 8; ++e) {
        int gi = i0 + wm + ti * 16 + p * 8 + e;
        Dp[(size_t)gi * SEQ + gj] = nAp[gi] + nbv - 2.0f * acc[ti][tj][e];
      }
    }
  }
}

// ------------- soft-DTW wavefront: 1 workgroup (512 thr) per batch -----------
__global__ __launch_bounds__(512) void softdtw_kernel(
    const float* __restrict__ D, float* __restrict__ Rout) {
  __shared__ float diag[3][SEQ + 1];   // rotating anti-diagonal buffers
  const int b = blockIdx.x;
  const int t = threadIdx.x;           // owns row i = t+1
  const float* Drow = D + ((size_t)b * SEQ + t) * SEQ;

  diag[0][t + 1] = BIGV;
  diag[1][t + 1] = BIGV;
  if (t == 0) { diag[0][0] = 0.0f; diag[1][0] = BIGV; diag[2][0] = BIGV; }
  __syncthreads();

  float4 dbuf = make_float4(0.f, 0.f, 0.f, 0.f);
  float v = BIGV;
  int cur = 2;  // buffer index == k % 3
  for (int k = 2; k <= 2 * SEQ; ++k) {
    int p1 = cur - 1; if (p1 < 0) p1 = 2;   // diag k-1
    int p2 = p1 - 1;  if (p2 < 0) p2 = 2;   // diag k-2
    int col = (k - 2) - t;                   // j-1 for this thread
    bool valid = ((unsigned)col < (unsigned)SEQ);
    if (valid && ((col & 3) == 0)) dbuf = *(const float4*)(Drow + col);
    int sel = col & 3;
    float d = (sel == 0) ? dbuf.x : (sel == 1) ? dbuf.y
            : (sel == 2) ? dbuf.z : dbuf.w;
    float a  = diag[p2][t];       // R[i-1][j-1]
    float bb = diag[p1][t];       // R[i-1][j]
    float cc = diag[p1][t + 1];   // R[i][j-1]
    float mn = fminf(a, fminf(bb, cc));
    float ssum = __expf((mn - a) * INV_GAMMA)
               + __expf((mn - bb) * INV_GAMMA)
               + __expf((mn - cc) * INV_GAMMA);
    float sm = mn - GAMMA_F * __logf(ssum);
    v = valid ? (d + sm) : BIGV;
    __syncthreads();                 // previous diag fully consumed
    diag[cur][t + 1] = v;
    if (t == 0) diag[cur][0] = BIGV;
    __syncthreads();                 // new diag visible
    cur = (cur == 2) ? 0 : cur + 1;
  }
  if (t == SEQ - 1) Rout[b] = v;     // R[N][M] at diagonal k = N+M
}

__global__ void dtw_combine_kernel(const float* __restrict__ R,
                                   float* __restrict__ out) {
  int b = threadIdx.x;
  if (b < BATCH)
    out[b] = R[b] - 0.5f * (R[BATCH + b] + R[2 * BATCH + b]);
}

extern "C" void kernel_launch(void* const* d_in, const int* in_sizes, int n_in,
                              void* d_out, int out_size, void* d_ws, size_t ws_size,
                              hipStream_t stream) {
  (void)in_sizes; (void)n_in; (void)out_size;
  const float* x = (const float*)d_in[0];
  const float* y = (const float*)d_in[1];
  float* out = (float*)d_out;
  char* ws = (char*)d_ws;

  const size_t dBytes    = (size_t)BATCH * SEQ * SEQ * sizeof(float);   // 64 MB
  const size_t normElems = (size_t)BATCH * SEQ;
  const size_t tailBytes = (2 * normElems + 3 * BATCH) * sizeof(float);

  // Triple-buffer D if workspace allows (more concurrent DTW blocks),
  // else reuse one buffer with stream-ordered gemm->dtw pairs.
  int nbuf = (ws_size >= 3 * dBytes + tailBytes) ? 3 : 1;

  float* Dbuf[3];
  Dbuf[0] = (float*)ws;
  Dbuf[1] = (nbuf == 3) ? (float*)(ws + dBytes) : Dbuf[0];
  Dbuf[2] = (nbuf == 3) ? (float*)(ws + 2 * dBytes) : Dbuf[0];
  float* nx = (float*)(ws + (size_t)nbuf * dBytes);
  float* ny = nx + normElems;
  float* R  = ny + normElems;

  {
    int waves  = 2 * BATCH * SEQ;  // one wave per row
    int blocks = waves / 8;        // 8 waves per 256-thread block
    hipLaunchKernelGGL(dtw_norms_kernel, dim3(blocks), dim3(256), 0, stream,
                       x, y, nx, ny);
  }

  const float* Am[3]  = {x, x, y};
  const float* Bm[3]  = {y, x, y};
  const float* nAm[3] = {nx, nx, ny};
  const float* nBm[3] = {ny, nx, ny};

  dim3 ggrid(SEQ / BT, SEQ / BT, BATCH);   // (4,4,64)
  if (nbuf == 3) {
    for (int m = 0; m < 3; ++m)
      hipLaunchKernelGGL(sqdist_wmma_kernel, ggrid, dim3(256), 0, stream,
                         Am[m], Bm[m], nAm[m], nBm[m], Dbuf[m]);
    for (int m = 0; m < 3; ++m)
      hipLaunchKernelGGL(softdtw_kernel, dim3(BATCH), dim3(SEQ), 0, stream,
                         Dbuf[m], R + m * BATCH);
  } else {
    for (int m = 0; m < 3; ++m) {
      hipLaunchKernelGGL(sqdist_wmma_kernel, ggrid, dim3(256), 0, stream,
                         Am[m], Bm[m], nAm[m], nBm[m], Dbuf[m]);
      hipLaunchKernelGGL(softdtw_kernel, dim3(BATCH), dim3(SEQ), 0, stream,
                         Dbuf[m], R + m * BATCH);
    }
  }
  hipLaunchKernelGGL(dtw_combine_kernel, dim3(1), dim3(BATCH), 0, stream, R, out);
}